// GMM2Mv1_58067957842069
// MI455X (gfx1250) — compile-verified
//
#include <hip/hip_runtime.h>
#include <hip/hip_bf16.h>

// ---------------- types / helpers ----------------
typedef __attribute__((ext_vector_type(2))) float v2f;
typedef __attribute__((ext_vector_type(8))) float v8f;

__device__ __forceinline__ v8f wmma4(v2f a, v2f b, v8f c) {
  // D = A(16x4 f32) * B(4x16 f32) + C(16x16 f32), exact fp32 matrix op
  return __builtin_amdgcn_wmma_f32_16x16x4_f32(false, a, false, b, (short)0, c,
                                               false, false);
}
__device__ __forceinline__ v8f vzero8() {
  v8f z = {0.f, 0.f, 0.f, 0.f, 0.f, 0.f, 0.f, 0.f};
  return z;
}
__device__ __forceinline__ float redmax16(float x) {
  x = fmaxf(x, __shfl_xor(x, 1, 32));
  x = fmaxf(x, __shfl_xor(x, 2, 32));
  x = fmaxf(x, __shfl_xor(x, 4, 32));
  x = fmaxf(x, __shfl_xor(x, 8, 32));
  return x;
}
__device__ __forceinline__ float redsum16(float x) {
  x += __shfl_xor(x, 1, 32);
  x += __shfl_xor(x, 2, 32);
  x += __shfl_xor(x, 4, 32);
  x += __shfl_xor(x, 8, 32);
  return x;
}

// gfx1250 async global->LDS copy (16B per lane), tracked by ASYNCcnt.
// LDS destination = low 32 bits of the flat shared pointer (LDS aperture
// keeps the byte offset in addr[31:0]).
__device__ __forceinline__ void async_g2l_16B(float* lds_dst,
                                              const float* gsrc) {
  unsigned ldsa = (unsigned)(unsigned long long)(uintptr_t)lds_dst;
  asm volatile("global_load_async_to_lds_b128 %0, %1, off"
               :
               : "v"(ldsa), "v"(gsrc)
               : "memory");
}
__device__ __forceinline__ void wait_async0() {
  asm volatile("s_wait_asynccnt 0x0" ::: "memory");
}

// ---------------- problem constants ----------------
constexpr int Bn = 2, Hh = 48, Ww = 96, Cc = 128;
constexpr int HWt = Hh * Ww;            // 4608
constexpr int WHh = 24, WWw = 48;       // window height/width (ns=2)
constexpr int Sw = WHh * WWw;           // 1152 tokens per window
constexpr float RSQRT_C = 0.088388347648318447f; // 1/sqrt(128)

// token t of window widx -> flattened (h,w) position, folding the ±(12,24) roll
__device__ __forceinline__ int win_pos(int widx, int t) {
  int s1 = widx >> 1, s2 = widx & 1;
  int yin = t / WWw, xin = t - yin * WWw;
  int y = s1 * WHh + yin + 12; if (y >= Hh) y -= Hh;
  int x = s2 * WWw + xin + 24; if (x >= Ww) x -= Ww;
  return y * Ww + x;
}

// =====================================================================
// Kernel 1: shifted-window flash attention.
// grid = 8 windows * 18 q-tiles(64 rows); block = 128 threads = 4 waves.
// =====================================================================
__global__ __launch_bounds__(128) void attn_kernel(
    const float* __restrict__ q, const float* __restrict__ k,
    const float* __restrict__ v, const float* __restrict__ mask,
    float* __restrict__ out) {
  constexpr int CP = 132;                  // padded row stride (floats)
  __shared__ float Qs[64 * CP];
  __shared__ float Ks[16 * CP];
  __shared__ float Vs[16 * CP];
  __shared__ float Ps[4 * 16 * 18];        // per-wave 16x16 P tile (pad 18)

  const int blk = blockIdx.x;
  const int wb = blk / 18;                 // window-batch 0..7
  const int mtile = blk % 18;
  const int bi = wb >> 2;
  const int widx = wb & 3;
  const int tid = threadIdx.x;
  const int wv = tid >> 5;
  const int lane = tid & 31;
  const int ln15 = lane & 15;
  const int half = lane >> 4;
  const int base = half * 2;               // K-offset for A/B VGPR layout
  const int rowbase = mtile * 64;

  const float* qb = q + (size_t)bi * HWt * Cc;
  const float* kb = k + (size_t)bi * HWt * Cc;
  const float* vb = v + (size_t)bi * HWt * Cc;
  const float* mb = mask + (size_t)widx * Sw * Sw;

  // stage Q tile (64 rows x 128 ch) via async DMA to LDS
  for (int idx = tid; idx < 64 * 32; idx += 128) {
    int r = idx >> 5, c4 = idx & 31;
    int pos = win_pos(widx, rowbase + r);
    async_g2l_16B(&Qs[r * CP + c4 * 4], qb + (size_t)pos * Cc + c4 * 4);
  }

  v8f acc[8];
  float mrow[8], lrow[8];
#pragma unroll
  for (int i = 0; i < 8; i++) { acc[i] = vzero8(); mrow[i] = -1e30f; lrow[i] = 0.f; }

  const int qrowA = wv * 16 + ln15;        // A-layout row (M = lane&15)
  float* PsW = &Ps[wv * 16 * 18];

  for (int nt = 0; nt < 72; ++nt) {
    __syncthreads();
    // stage K/V tile (16 keys x 128 ch) via async DMA
    for (int idx = tid; idx < 16 * 32; idx += 128) {
      int r = idx >> 5, c4 = idx & 31;
      int pos = win_pos(widx, nt * 16 + r);
      async_g2l_16B(&Ks[r * CP + c4 * 4], kb + (size_t)pos * Cc + c4 * 4);
      async_g2l_16B(&Vs[r * CP + c4 * 4], vb + (size_t)pos * Cc + c4 * 4);
    }
    wait_async0();               // also covers the Q staging on iter 0
    __syncthreads();

    // S = Q * K^T  (16x16 tile, K-dim 128 => 32 wmma steps)
    v8f s = vzero8();
#pragma unroll
    for (int kk = 0; kk < 32; ++kk) {
      v2f a, bvec;
      a.x = Qs[qrowA * CP + kk * 4 + base];
      a.y = Qs[qrowA * CP + kk * 4 + base + 1];
      bvec.x = Ks[ln15 * CP + kk * 4 + base];
      bvec.y = Ks[ln15 * CP + kk * 4 + base + 1];
      s = wmma4(a, bvec, s);
    }

    // scale + shift-window mask
    const int jcol = nt * 16 + ln15;
#pragma unroll
    for (int r = 0; r < 8; r++) {
      int row_d = rowbase + wv * 16 + half * 8 + r;
      s[r] = s[r] * RSQRT_C + mb[(size_t)row_d * Sw + jcol];
    }

    // online softmax update (rows live in 16-lane halves of the wave)
    float p[8], cf[8];
#pragma unroll
    for (int r = 0; r < 8; r++) {
      float tm = redmax16(s[r]);
      float mnew = fmaxf(mrow[r], tm);
      float c = __expf(mrow[r] - mnew);
      float pv = __expf(s[r] - mnew);
      float ps = redsum16(pv);
      lrow[r] = lrow[r] * c + ps;
      mrow[r] = mnew;
      cf[r] = c;
      p[r] = pv;
    }
#pragma unroll
    for (int ct = 0; ct < 8; ++ct)
#pragma unroll
      for (int r = 0; r < 8; r++) acc[ct][r] *= cf[r];

    // D-layout -> A-layout via per-wave LDS tile
#pragma unroll
    for (int r = 0; r < 8; r++) PsW[(half * 8 + r) * 18 + ln15] = p[r];
    __builtin_amdgcn_wave_barrier();

    // O += P * V   (M=16, N=128, K=16)
#pragma unroll
    for (int ct = 0; ct < 8; ++ct) {
      v8f o = acc[ct];
#pragma unroll
      for (int kk = 0; kk < 4; ++kk) {
        v2f a, bvec;
        a.x = PsW[ln15 * 18 + kk * 4 + base];
        a.y = PsW[ln15 * 18 + kk * 4 + base + 1];
        bvec.x = Vs[(kk * 4 + base) * CP + ct * 16 + ln15];
        bvec.y = Vs[(kk * 4 + base + 1) * CP + ct * 16 + ln15];
        o = wmma4(a, bvec, o);
      }
      acc[ct] = o;
    }
  }

  // epilogue: normalize, stage result tile in LDS (reuse Qs), then write
  // full 512B-contiguous rows through the inverse roll/merge mapping.
  __syncthreads();                       // everyone is done reading Qs
#pragma unroll
  for (int r = 0; r < 8; r++) {
    float inv = 1.f / lrow[r];
    int rloc = wv * 16 + half * 8 + r;
#pragma unroll
    for (int ct = 0; ct < 8; ct++)
      Qs[rloc * CP + ct * 16 + ln15] = acc[ct][r] * inv;
  }
  __syncthreads();
  for (int idx = tid; idx < 64 * 32; idx += 128) {
    int r = idx >> 5, c4 = idx & 31;
    int pos = win_pos(widx, rowbase + r);
    *reinterpret_cast<float4*>(out + ((size_t)bi * HWt + pos) * Cc + c4 * 4) =
        *reinterpret_cast<const float4*>(&Qs[r * CP + c4 * 4]);
  }
}

// =====================================================================
// Kernel 2: softmax stats over corr rows (or corrT rows when F0/F1 are
// swapped) + grid-weighted sums -> flow output. Never materializes corr.
// grid = B * 72 (64-row tiles); block = 128 threads = 4 waves.
// =====================================================================
__global__ __launch_bounds__(128) void corr_stats_kernel(
    const float* __restrict__ F0, const float* __restrict__ F1,
    float* __restrict__ stat_m, float* __restrict__ stat_l,
    float* __restrict__ flow) {
  constexpr int IP = 66;  // padded i-stride
  constexpr int JP = 18;  // padded j-stride
  __shared__ float As[128 * IP];  // [c][i] for 64 rows
  __shared__ float Bs[128 * JP];  // [c][j] for 16 cols

  const int blk = blockIdx.x;
  const int bi = blk / 72;
  const int i0 = (blk % 72) * 64;
  const int tid = threadIdx.x;
  const int wv = tid >> 5;
  const int lane = tid & 31;
  const int ln15 = lane & 15;
  const int half = lane >> 4;
  const int base = half * 2;

  const float* f0b = F0 + (size_t)bi * Cc * HWt;
  const float* f1b = F1 + (size_t)bi * Cc * HWt;

  // stage A tile: 128 c x 64 i (c-major, contiguous in i) via async DMA
  for (int idx = tid; idx < 128 * 16; idx += 128) {
    int c = idx >> 4, i4 = idx & 15;
    async_g2l_16B(&As[c * IP + i4 * 4], f0b + (size_t)c * HWt + i0 + i4 * 4);
  }

  float mrow[8], lrow[8], sx[8], sy[8];
#pragma unroll
  for (int r = 0; r < 8; r++) { mrow[r] = -1e30f; lrow[r] = 0.f; sx[r] = 0.f; sy[r] = 0.f; }

  for (int jt = 0; jt < 288; ++jt) {
    __syncthreads();
    for (int idx = tid; idx < 128 * 4; idx += 128) {
      int c = idx >> 2, j4 = idx & 3;
      async_g2l_16B(&Bs[c * JP + j4 * 4],
                    f1b + (size_t)c * HWt + jt * 16 + j4 * 4);
    }
    wait_async0();
    __syncthreads();

    v8f s = vzero8();
#pragma unroll
    for (int kk = 0; kk < 32; ++kk) {
      v2f a, bvec;
      a.x = As[(kk * 4 + base) * IP + wv * 16 + ln15];
      a.y = As[(kk * 4 + base + 1) * IP + wv * 16 + ln15];
      bvec.x = Bs[(kk * 4 + base) * JP + ln15];
      bvec.y = Bs[(kk * 4 + base + 1) * JP + ln15];
      s = wmma4(a, bvec, s);
    }

    const int j = jt * 16 + ln15;
    const float xj = (float)(j % Ww), yj = (float)(j / Ww);
#pragma unroll
    for (int r = 0; r < 8; r++) {
      float sv = s[r] * RSQRT_C;
      float tm = redmax16(sv);
      float mnew = fmaxf(mrow[r], tm);
      float c = __expf(mrow[r] - mnew);
      float pv = __expf(sv - mnew);
      float t0 = redsum16(pv);
      float t1 = redsum16(pv * xj);
      float t2 = redsum16(pv * yj);
      lrow[r] = lrow[r] * c + t0;
      sx[r] = sx[r] * c + t1;
      sy[r] = sy[r] * c + t2;
      mrow[r] = mnew;
    }
  }

  if (ln15 == 0) {
#pragma unroll
    for (int r = 0; r < 8; r++) {
      int i = i0 + wv * 16 + half * 8 + r;
      stat_m[bi * HWt + i] = mrow[r];
      stat_l[bi * HWt + i] = lrow[r];
      float xi = (float)(i % Ww), yi = (float)(i / Ww);
      float inv = 1.f / lrow[r];
      flow[((size_t)bi * 2 + 0) * HWt + i] = sx[r] * inv - xi;
      flow[((size_t)bi * 2 + 1) * HWt + i] = sy[r] * inv - yi;
    }
  }
}

// =====================================================================
// Kernel 3: dual_prob tiles. Recompute corr 64x32 tile with WMMA, apply
// dual = exp(2c - rm_i - cm_j) / (rl_i * cl_j), stage in LDS, stream out
// as 128B-contiguous rows. grid = B * 72 * 144; block = 128 = 4 waves.
// =====================================================================
__global__ __launch_bounds__(128) void dual_kernel(
    const float* __restrict__ F0, const float* __restrict__ F1,
    const float* __restrict__ rm, const float* __restrict__ rl,
    const float* __restrict__ cm, const float* __restrict__ cl,
    float* __restrict__ dual) {
  constexpr int IP = 66;
  constexpr int JP = 34;
  constexpr int OP = 36;          // out-staging stride (floats)
  __shared__ float As[128 * IP];  // [c][i], 64 rows (reused as out staging)
  __shared__ float Bs[128 * JP];  // [c][j], 32 cols

  const int blk = blockIdx.x;
  const int bi = blk / (72 * 144);
  const int rem = blk % (72 * 144);
  const int i0 = (rem / 144) * 64;
  const int j0 = (rem % 144) * 32;
  const int tid = threadIdx.x;
  const int wv = tid >> 5;
  const int lane = tid & 31;
  const int ln15 = lane & 15;
  const int half = lane >> 4;
  const int base = half * 2;

  const float* f0b = F0 + (size_t)bi * Cc * HWt;
  const float* f1b = F1 + (size_t)bi * Cc * HWt;

  for (int idx = tid; idx < 128 * 16; idx += 128) {
    int c = idx >> 4, i4 = idx & 15;
    async_g2l_16B(&As[c * IP + i4 * 4], f0b + (size_t)c * HWt + i0 + i4 * 4);
  }
  for (int idx = tid; idx < 128 * 8; idx += 128) {
    int c = idx >> 3, j4 = idx & 7;
    async_g2l_16B(&Bs[c * JP + j4 * 4], f1b + (size_t)c * HWt + j0 + j4 * 4);
  }
  wait_async0();
  __syncthreads();

  float rmv[8], rlv[8];
#pragma unroll
  for (int r = 0; r < 8; r++) {
    int i = i0 + wv * 16 + half * 8 + r;
    rmv[r] = rm[bi * HWt + i];
    rlv[r] = rl[bi * HWt + i];
  }

  v8f s[2];
#pragma unroll
  for (int ntl = 0; ntl < 2; ++ntl) {
    v8f acc = vzero8();
#pragma unroll
    for (int kk = 0; kk < 32; ++kk) {
      v2f a, bvec;
      a.x = As[(kk * 4 + base) * IP + wv * 16 + ln15];
      a.y = As[(kk * 4 + base + 1) * IP + wv * 16 + ln15];
      bvec.x = Bs[(kk * 4 + base) * JP + ntl * 16 + ln15];
      bvec.y = Bs[(kk * 4 + base + 1) * JP + ntl * 16 + ln15];
      acc = wmma4(a, bvec, acc);
    }
    s[ntl] = acc;
  }

  __syncthreads();          // all waves done reading As/Bs; reuse As
  float* Cs = As;
#pragma unroll
  for (int ntl = 0; ntl < 2; ++ntl) {
    const int j = j0 + ntl * 16 + ln15;
    const float cmv = cm[bi * HWt + j];
    const float clinv = 1.f / cl[bi * HWt + j];
#pragma unroll
    for (int r = 0; r < 8; r++) {
      float val = __expf(2.f * (s[ntl][r] * RSQRT_C) - rmv[r] - cmv) *
                  (clinv / rlv[r]);
      Cs[(wv * 16 + half * 8 + r) * OP + ntl * 16 + ln15] = val;
    }
  }
  __syncthreads();

  // contiguous 128B row segments per store
  for (int idx = tid; idx < 64 * 8; idx += 128) {
    int row = idx >> 3, j4 = idx & 7;
    *reinterpret_cast<float4*>(dual + ((size_t)bi * HWt + i0 + row) * HWt +
                               j0 + j4 * 4) =
        *reinterpret_cast<const float4*>(&Cs[row * OP + j4 * 4]);
  }
}

// ---------------- host launcher ----------------
extern "C" void kernel_launch(void* const* d_in, const int* in_sizes, int n_in,
                              void* d_out, int out_size, void* d_ws,
                              size_t ws_size, hipStream_t stream) {
  (void)in_sizes; (void)n_in; (void)out_size; (void)ws_size;
  const float* q = (const float*)d_in[0];
  const float* k = (const float*)d_in[1];
  const float* v = (const float*)d_in[2];
  const float* f0 = (const float*)d_in[3];
  const float* f1 = (const float*)d_in[4];
  const float* mask = (const float*)d_in[5];
  float* out = (float*)d_out;
  float* ws = (float*)d_ws;

  // output layout (concatenated): attn (2*4608*128), f01 (2*2*4608),
  // f10 (2*2*4608), dual (2*4608*4608)
  float* attn_out = out;
  float* f01_out = out + 1179648;
  float* f10_out = out + 1198080;
  float* dual_out = out + 1216512;

  float* rm = ws;           // [2][4608]
  float* rl = ws + 9216;
  float* cmv = ws + 18432;
  float* clv = ws + 27648;

  attn_kernel<<<144, 128, 0, stream>>>(q, k, v, mask, attn_out);
  corr_stats_kernel<<<144, 128, 0, stream>>>(f0, f1, rm, rl, f01_out);
  corr_stats_kernel<<<144, 128, 0, stream>>>(f1, f0, cmv, clv, f10_out);
  dual_kernel<<<20736, 128, 0, stream>>>(f0, f1, rm, rl, cmv, clv, dual_out);
}